// GaussianMomentDescriptor_86827058856482
// MI455X (gfx1250) — compile-verified
//
#include <hip/hip_runtime.h>
#include <hip/hip_bf16.h>
#include <math.h>

#define NATOMS   10000
#define NPAIRS   600000
#define NBASIS   7
#define NRADIAL  5
#define NSPECIES 119
#define RMINF    0.5f
#define RMAXF    6.0f
#define PI_F     3.14159265358979f

typedef __attribute__((ext_vector_type(2))) float v2f;
typedef __attribute__((ext_vector_type(8))) float v8f;

// symmetric-index tables: unique index for sorted (a,b) / (a,b,c) over dim 3
__constant__ int c_T2[9]  = {0,1,2, 1,3,4, 2,4,5};
__constant__ int c_T3[27] = {0,1,2,1,3,4,2,4,5, 1,3,4,3,6,7,4,7,8, 2,4,5,4,7,8,5,8,9};
// triu pairs (r<=s) over 5, row-major (np.triu_indices order)
__constant__ int c_PR[15] = {0,0,0,0,0,1,1,1,1,2,2,2,3,3,4};
__constant__ int c_PS[15] = {0,1,2,3,4,1,2,3,4,2,3,4,3,4,4};
// sorted triples r<=s<=t over 5 (35), loop order a,b>=a,c>=b
__constant__ int c_TR[35] = {0,0,0,0,0,0,0,0,0,0,0,0,0,0,0, 1,1,1,1,1,1,1,1,1,1, 2,2,2,2,2,2, 3,3,3, 4};
__constant__ int c_TS[35] = {0,0,0,0,0,1,1,1,1,2,2,2,3,3,4, 1,1,1,1,2,2,2,3,3,4, 2,2,2,3,3,4, 3,3,4, 4};
__constant__ int c_TT[35] = {0,1,2,3,4,1,2,3,4,2,3,4,3,4,4, 1,2,3,4,2,3,4,3,4,4, 2,3,4,3,4,4, 3,4,4, 4};

// ---------------------------------------------------------------------------
// workspace zero-fill (moment accumulators must be 0 at every launch)
// ---------------------------------------------------------------------------
__global__ void __launch_bounds__(256) gm_zero_kernel(float* __restrict__ p, int n) {
  int t = blockIdx.x * blockDim.x + threadIdx.x;
  if (t < n) p[t] = 0.0f;
}

// ---------------------------------------------------------------------------
// phase 1: per-pair radial basis + symmetric-unique moment scatter (L2 atomics)
// unique layout per atom (100 f32): [0,5)=m0  [5,20)=m1  [20,50)=m2u  [50,100)=m3u
// ---------------------------------------------------------------------------
__global__ void __launch_bounds__(256) gm_pair_kernel(
    const float* __restrict__ R, const int* __restrict__ Z,
    const int* __restrict__ idx, const float* __restrict__ emb,
    float* __restrict__ mom)
{
  int p = blockIdx.x * blockDim.x + threadIdx.x;
  if (p >= NPAIRS) return;

  const int ai = idx[p];
  const int aj = idx[NPAIRS + p];

  // species-pair coefficient row (2 MB table, L2-resident) — prefetch early
  const float* cf = &emb[(((size_t)Z[aj] * NSPECIES) + (size_t)Z[ai]) * (NRADIAL * NBASIS)];
  __builtin_prefetch(cf, 0, 0);   // -> global_prefetch_b8

  float dx = R[3*ai+0] - R[3*aj+0];
  float dy = R[3*ai+1] - R[3*aj+1];
  float dz = R[3*ai+2] - R[3*aj+2];
  float dr  = sqrtf(dx*dx + dy*dy + dz*dz);
  float inv = 1.0f / dr;

  const float betta      = (float)(NBASIS*NBASIS) / (RMAXF*RMAXF); // 49/36
  const float rad_norm   = 0.96479213f;   // (2*betta/pi)^0.25
  const float embed_norm = 0.37796447f;   // 1/sqrt(7)

  float drc    = fminf(dr, RMAXF);
  float cutoff = 0.5f * (__cosf(PI_F * drc / RMAXF) + 1.0f);

  float basis[NBASIS];
#pragma unroll
  for (int b = 0; b < NBASIS; ++b) {
    float sh = RMINF + (RMAXF - RMINF) / (float)NBASIS * (float)b - dr;
    basis[b] = rad_norm * __expf(-betta * sh * sh);
  }

  const float scale = embed_norm * cutoff;
  float rad[NRADIAL];
#pragma unroll
  for (int r = 0; r < NRADIAL; ++r) {
    float acc = 0.0f;
#pragma unroll
    for (int b = 0; b < NBASIS; ++b) acc += cf[r*NBASIS + b] * basis[b];
    rad[r] = acc * scale;
  }

  float nx = dx*inv, ny = dy*inv, nz = dz*inv;
  // unique 2nd-order: (00)(01)(02)(11)(12)(22)
  float q[6] = {nx*nx, nx*ny, nx*nz, ny*ny, ny*nz, nz*nz};
  // unique 3rd-order: (000)(001)(002)(011)(012)(022)(111)(112)(122)(222)
  float t3[10] = {q[0]*nx, q[0]*ny, q[0]*nz, q[3]*nx, q[4]*nx,
                  q[5]*nx, q[3]*ny, q[3]*nz, q[5]*ny, q[5]*nz};

  float* base = mom + (size_t)ai * 100;
#pragma unroll
  for (int r = 0; r < NRADIAL; ++r) {
    float rv = rad[r];
    unsafeAtomicAdd(base + r, rv);
    unsafeAtomicAdd(base + 5 + r*3 + 0, rv*nx);
    unsafeAtomicAdd(base + 5 + r*3 + 1, rv*ny);
    unsafeAtomicAdd(base + 5 + r*3 + 2, rv*nz);
#pragma unroll
    for (int u = 0; u < 6; ++u)  unsafeAtomicAdd(base + 20 + r*6  + u, rv*q[u]);
#pragma unroll
    for (int u = 0; u < 10; ++u) unsafeAtomicAdd(base + 50 + r*10 + u, rv*t3[u]);
  }
}

// ---------------------------------------------------------------------------
// V_WMMA_F32_16X16X4_F32 helper (ISA 7.12.2 layouts)
// A 16x4: lane=M; VGPR0/1 = K {0,1} (lanes 0-15) / K {2,3} (lanes 16-31)
// B 4x16: lane=N; VGPR0/1 = K {0,1} (lanes 0-15) / K {2,3} (lanes 16-31)
// C/D 16x16: VGPR g = row g (lanes 0-15) / row g+8 (lanes 16-31), col = lane&15
// Operands are pre-staged in LDS as zero-padded 16x12 (row-major for A,
// column-major for B) so a fragment = one aligned ds_load_b64, branchless.
// ---------------------------------------------------------------------------
__device__ inline v8f wmma4(v2f a, v2f b, v8f c) {
  return __builtin_amdgcn_wmma_f32_16x16x4_f32(false, a, false, b, (short)0, c, false, false);
}

// ---------------------------------------------------------------------------
// phase 2: per-atom contractions, one atom per wave32, fp32 WMMA
// output layout: c0@0(5) c1@5(15) c2@20(15) c3@35(15) c4@50(35)
//                c5@85(75) c6@160(75) c7@235(125)  -> 360
// ---------------------------------------------------------------------------
__global__ void __launch_bounds__(128) gm_atom_kernel(const float* __restrict__ mom,
                                                      float* __restrict__ out)
{
  __shared__ float lds[4][1288];
  const int wave = threadIdx.x >> 5;
  const int lane = threadIdx.x & 31;
  const int atom = blockIdx.x * 4 + wave;        // grid = NATOMS/4 exactly
  float* M  = lds[wave] + 0;     // 200: expanded moments
  float* S  = lds[wave] + 200;   // 256: 16x16 staging (T / TM / U)
  float* A0 = lds[wave] + 456;   // 192: 16x12 m3 as [(r,k)][(i,j)]  (A of T/TM; B of T)
  float* B1 = lds[wave] + 648;   // 192: 16x12 m2^T as [t][(i,j)]    (B of c5/c6/c7)
  float* A1 = lds[wave] + 840;   // 192: 16x12 m1-outer [rs][(i,j)]  (A of c5)
  float* AT = lds[wave] + 1032;  // 192: 16x12 T-pairs [rs][(k,l)]   (A of c6)
  float* AU = lds[wave] + 1224;  // 64:  16x4  m2 as [(r,j)][i]      (A and B of U)
  const float* g = mom + (size_t)atom * 100;
  float* O = out + (size_t)atom * 360;

  // expand unique -> full tensors in LDS: [0,5)m0 [5,20)m1 [20,65)m2 [65,200)m3
  for (int t = lane; t < 200; t += 32) {
    int src;
    if (t < 20) {
      src = t;
    } else if (t < 65) {
      int z = t - 20, r = z / 9, ij = z % 9, i = ij / 3, j = ij % 3;
      int a = i < j ? i : j, b = i < j ? j : i;
      src = 20 + r*6 + c_T2[a*3 + b];
    } else {
      int z = t - 65, r = z / 27, rem = z % 27;
      src = 50 + r*10 + c_T3[rem];
    }
    M[t] = g[src];
  }
  // zero the contiguous operand-staging block A0..AU (832 floats)
  for (int t = lane; t < 832; t += 32) A0[t] = 0.0f;
  __syncthreads();

  auto m1v = [&](int r, int i){ return M[5 + r*3 + i]; };
  auto m2v = [&](int tt, int i, int j){ return M[20 + tt*9 + i*3 + j]; };
  auto m3v = [&](int r, int i, int j, int k){ return M[65 + r*27 + i*9 + j*3 + k]; };

  // fill operand tiles (padding stays zero) — branchless strided loops
  for (int e = lane; e < 135; e += 32) {
    int rk = e / 9, ij = e % 9;
    A0[rk*12 + ij] = m3v(rk/3, ij/3, ij%3, rk%3);
    A1[rk*12 + ij] = m1v(c_PR[rk], ij/3) * m1v(c_PS[rk], ij%3);
  }
  for (int e = lane; e < 45; e += 32) {
    int t = e / 9, ij = e % 9;
    B1[t*12 + ij] = m2v(t, ij/3, ij%3);
    int rj = e / 3, i = e % 3;
    AU[rj*4 + i] = m2v(rj/3, i, rj%3);
  }
  __syncthreads();

  const int n  = lane & 15, hi = lane >> 4;
  const int fo  = n*12 + 2*hi;   // fragment offset into 16x12 tiles (even -> b64)
  const int fo4 = n*4  + 2*hi;   // fragment offset into 16x4 tile

  // ---- T[(r,k),(s,l)] = sum_ij m3[r,ij,k]*m3[s,ij,l]
  //      A and (column-major) B fragments are the SAME registers.
  v8f T = {};
#pragma unroll
  for (int c = 0; c < 3; ++c) {
    v2f ab = *(const v2f*)(A0 + fo + c*4);
    T = wmma4(ab, ab, T);
  }
#pragma unroll
  for (int gi = 0; gi < 8; ++gi) S[(gi + 8*hi)*16 + n] = T[gi];
  __syncthreads();

  // c3[r,s] = sum_k T[(r,k),(s,k)]
  if (lane < 15) {
    int r = c_PR[lane], s = c_PS[lane];
    float v = 0.0f;
#pragma unroll
    for (int k = 0; k < 3; ++k) v += S[(r*3 + k)*16 + s*3 + k];
    O[35 + lane] = v;
  }
  // gather A of c6 from staged T
  for (int e = lane; e < 135; e += 32) {
    int rs = e / 9, kl = e % 9;
    AT[rs*12 + kl] = S[(c_PR[rs]*3 + kl/3)*16 + c_PS[rs]*3 + kl%3];
  }
  __syncthreads();

  // c6 = AT @ m2^T ; c5 = A1 @ m2^T ; TM = A0 @ m2^T (for c7)
  v8f C6 = {}, C5 = {}, TM = {};
#pragma unroll
  for (int c = 0; c < 3; ++c) {
    v2f b = *(const v2f*)(B1 + fo + c*4);
    C6 = wmma4(*(const v2f*)(AT + fo + c*4), b, C6);
    C5 = wmma4(*(const v2f*)(A1 + fo + c*4), b, C5);
    TM = wmma4(*(const v2f*)(A0 + fo + c*4), b, TM);
  }
  __syncthreads();                       // all readers of T in S are done
#pragma unroll
  for (int gi = 0; gi < 8; ++gi) S[(gi + 8*hi)*16 + n] = TM[gi];
  __syncthreads();

  // store c5 (85) / c6 (160): valid 15x5 tile, index = pair*5 + t
#pragma unroll
  for (int gi = 0; gi < 8; ++gi) {
    int row = gi + 8*hi;
    if (row < 15 && n < 5) {
      O[85  + row*5 + n] = C5[gi];
      O[160 + row*5 + n] = C6[gi];
    }
  }

  // c7[r,s,t] = sum_k TM[(r,k),s] * m1[t,k]   (125 values)
  for (int q = lane; q < 125; q += 32) {
    int r = q / 25, s = (q / 5) % 5, t = q % 5;
    float v = 0.0f;
#pragma unroll
    for (int k = 0; k < 3; ++k) v += S[(r*3 + k)*16 + s] * m1v(t, k);
    O[235 + q] = v;
  }
  __syncthreads();                       // c7 readers of S done

  // U[(r,j),(s,k)] = sum_i m2[r,i,j]*m2[s,i,k] — A==B fragment again
  {
    v2f au = *(const v2f*)(AU + fo4);
    v8f U = wmma4(au, au, (v8f){});
#pragma unroll
    for (int gi = 0; gi < 8; ++gi) S[(gi + 8*hi)*16 + n] = U[gi];
  }
  __syncthreads();

  // c4[r,s,t] = sum_jk U[(r,j),(s,k)] * m2[t,j,k]   (35 sorted triples)
  for (int q = lane; q < 35; q += 32) {
    int r = c_TR[q], s = c_TS[q], t = c_TT[q];
    float v = 0.0f;
#pragma unroll
    for (int jk = 0; jk < 9; ++jk)
      v += S[(r*3 + jk/3)*16 + s*3 + jk%3] * m2v(t, jk/3, jk%3);
    O[50 + q] = v;
  }

  // c0, c1, c2
  if (lane < 5) O[lane] = M[lane];
  if (lane < 15) {
    int r = c_PR[lane], s = c_PS[lane];
    float v1 = 0.0f, v2 = 0.0f;
#pragma unroll
    for (int i = 0; i < 3; ++i)  v1 += m1v(r, i) * m1v(s, i);
#pragma unroll
    for (int ij = 0; ij < 9; ++ij) v2 += M[20 + r*9 + ij] * M[20 + s*9 + ij];
    O[5 + lane]  = v1;
    O[20 + lane] = v2;
  }
}

// ---------------------------------------------------------------------------
extern "C" void kernel_launch(void* const* d_in, const int* in_sizes, int n_in,
                              void* d_out, int out_size, void* d_ws, size_t ws_size,
                              hipStream_t stream) {
  (void)in_sizes; (void)n_in; (void)out_size; (void)ws_size;
  const float* R   = (const float*)d_in[0];
  const int*   Z   = (const int*)d_in[1];
  const int*   idx = (const int*)d_in[2];
  const float* emb = (const float*)d_in[3];
  float* out = (float*)d_out;
  float* mom = (float*)d_ws;               // NATOMS*100 f32 = 4 MB scratch

  const int nz = NATOMS * 100;
  gm_zero_kernel<<<(nz + 255) / 256, 256, 0, stream>>>(mom, nz);
  gm_pair_kernel<<<(NPAIRS + 255) / 256, 256, 0, stream>>>(R, Z, idx, emb, mom);
  gm_atom_kernel<<<NATOMS / 4, 128, 0, stream>>>(mom, out);
}